// PrototypeLayer_58497454572031
// MI455X (gfx1250) — compile-verified
//
#include <hip/hip_runtime.h>

typedef __attribute__((ext_vector_type(2))) float v2f;
typedef __attribute__((ext_vector_type(8))) float v8f;

#define B_SZ 512
#define P_SZ 200
#define D_SZ 512
#define SMALLD 16
#define CLAMP_EPS 1e-6f
#define NSWEEP 8

__global__ __launch_bounds__(256) void grassmann_dist_kernel(
    const float* __restrict__ xs,       // (B, D, d) row-major
    const float* __restrict__ xprotos,  // (P, D, d) row-major
    const float* __restrict__ rel,      // (d,)
    float* __restrict__ out)            // (B, P)
{
  // LDS: [0,8192) X-tile (512x16 f32), [8192,10240) per-wave M (8 x 256),
  //      [10240,10368) per-wave singular values (8 x 16)
  __shared__ float smem[8192 + 2048 + 128];

  const int tid   = threadIdx.x;
  const int lane  = tid & 31;
  const int wave  = tid >> 5;
  const int b     = blockIdx.y;
  const int proto = blockIdx.x * 8 + wave;

  // ---- Stage X_b (512x16 fp32 = 32KB) into LDS cooperatively (float4) ----
  {
    const float4* Xb4 = (const float4*)(xs + (size_t)b * (D_SZ * SMALLD));
    float4* S4 = (float4*)smem;
#pragma unroll
    for (int i = 0; i < 8; ++i)
      S4[tid + i * 256] = Xb4[tid + i * 256];
  }
  __syncthreads();

  // ---- Gram M = X_b^T (16x512) @ Y_p (512x16) via V_WMMA_F32_16X16X4_F32 ----
  // A-frag (16x4): lanes 0-15 hold K=0,1 in v0,v1 ; lanes 16-31 hold K=2,3.
  // B-frag (4x16): same K split across lane halves; N = lane&15.
  const float* Yp   = xprotos + (size_t)proto * (D_SZ * SMALLD);
  const int kbase   = (lane >> 4) * 2;  // 0 or 2
  const int colc    = lane & 15;
  v8f acc = {0.f, 0.f, 0.f, 0.f, 0.f, 0.f, 0.f, 0.f};
#pragma unroll 4
  for (int c = 0; c < D_SZ; c += 4) {
    const float* xr = &smem[(c + kbase) * SMALLD + colc];
    const float* yr = &Yp[(size_t)(c + kbase) * SMALLD + colc];
    v2f a, bb;
    a[0]  = xr[0];  a[1]  = xr[SMALLD];
    bb[0] = yr[0];  bb[1] = yr[SMALLD];
    acc = __builtin_amdgcn_wmma_f32_16x16x4_f32(
        /*neg_a=*/false, a, /*neg_b=*/false, bb,
        /*c_mod=*/(short)0, acc, /*reuse_a=*/false, /*reuse_b=*/false);
  }

  // ---- Spill M (16x16) to LDS, column-major: Mw[col*16 + row] ----
  // C layout: v[i] = M[row0+i, lane&15], row0 = 8*(lane>>4)
  float* Mw = &smem[8192 + wave * 256];
  {
    const int r0 = (lane >> 4) * 8;
#pragma unroll
    for (int i = 0; i < 8; ++i)
      Mw[colc * 16 + r0 + i] = acc[i];
  }
  __builtin_amdgcn_wave_barrier();

  // ---- One-sided (Hestenes) Jacobi: orthogonalize the 16 columns of M ----
  // 8 pairs per round, 4 lanes per pair (each lane owns 4 of 16 elements).
  const int g   = lane >> 2;  // pair id 0..7
  const int sub = lane & 3;   // element quarter
  for (int sweep = 0; sweep < NSWEEP; ++sweep) {
    for (int r = 0; r < 15; ++r) {
      // round-robin tournament pairing (player 0 fixed)
      const int pp = (g == 0) ? 0 : (1 + (g - 1 + r) % 15);
      const int qq = 1 + (14 - g + r) % 15;
      float xp[4], xq[4];
      float al = 0.f, be = 0.f, ga = 0.f;
#pragma unroll
      for (int e = 0; e < 4; ++e) {
        xp[e] = Mw[pp * 16 + sub * 4 + e];
        xq[e] = Mw[qq * 16 + sub * 4 + e];
        al += xp[e] * xp[e];
        be += xq[e] * xq[e];
        ga += xp[e] * xq[e];
      }
      // reduce partial dots across the 4 lanes of this pair group
      al += __shfl_xor(al, 1); al += __shfl_xor(al, 2);
      be += __shfl_xor(be, 1); be += __shfl_xor(be, 2);
      ga += __shfl_xor(ga, 1); ga += __shfl_xor(ga, 2);
      const bool  rot  = fabsf(ga) > 1e-12f;
      const float zeta = rot ? (be - al) / (2.f * ga) : 0.f;
      const float t    = rot ? copysignf(1.f, zeta) /
                               (fabsf(zeta) + sqrtf(1.f + zeta * zeta))
                             : 0.f;
      const float cth = 1.f / sqrtf(1.f + t * t);
      const float sth = cth * t;
#pragma unroll
      for (int e = 0; e < 4; ++e) {
        const float np = cth * xp[e] - sth * xq[e];
        const float nq = sth * xp[e] + cth * xq[e];
        Mw[pp * 16 + sub * 4 + e] = np;
        Mw[qq * 16 + sub * 4 + e] = nq;
      }
      __builtin_amdgcn_wave_barrier();
    }
  }

  // ---- Singular values = column norms; rank descending; weighted acos^2 ----
  if (lane < 16) {
    float n2 = 0.f;
#pragma unroll
    for (int r = 0; r < 16; ++r) {
      const float v = Mw[lane * 16 + r];
      n2 += v * v;
    }
    const float sv = sqrtf(n2);
    float* Sw = &smem[8192 + 2048 + wave * 16];
    Sw[lane] = sv;
    __builtin_amdgcn_wave_barrier();
    int rank = 0;
#pragma unroll
    for (int i = 0; i < 16; ++i) {
      const float si = Sw[i];
      rank += (si > sv) || ((si == sv) && (i < lane));
    }
    const float sc = fminf(fmaxf(sv, -1.f + CLAMP_EPS), 1.f - CLAMP_EPS);
    const float th = acosf(sc);
    float contrib = rel[rank] * th * th;
    contrib += __shfl_xor(contrib, 1);
    contrib += __shfl_xor(contrib, 2);
    contrib += __shfl_xor(contrib, 4);
    contrib += __shfl_xor(contrib, 8);
    if (lane == 0) out[(size_t)b * P_SZ + proto] = contrib;
  }
}

extern "C" void kernel_launch(void* const* d_in, const int* in_sizes, int n_in,
                              void* d_out, int out_size, void* d_ws, size_t ws_size,
                              hipStream_t stream) {
  (void)in_sizes; (void)n_in; (void)out_size; (void)d_ws; (void)ws_size;
  const float* xs   = (const float*)d_in[0];  // (512, 512, 16) fp32
  const float* xpro = (const float*)d_in[1];  // (200, 512, 16) fp32
  const float* rel  = (const float*)d_in[2];  // (1, 16) fp32
  float* out = (float*)d_out;                 // (512, 200) fp32

  dim3 grid(P_SZ / 8, B_SZ, 1);  // 25 x 512 blocks, 8 waves -> 8 prototypes each
  grassmann_dist_kernel<<<grid, 256, 0, stream>>>(xs, xpro, rel, out);
}